// DiffeoDeformedMSSField_64390149701964
// MI455X (gfx1250) — compile-verified
//
#include <hip/hip_runtime.h>

// MI455X / gfx1250: wave32 WMMA f32 16x16x4 for the batched 3->64->3 neural-ODE
// MLP. Operand-swapped (weights=A, data=B) keeps state point-along-lanes, and a
// custom K-permutation of the layer-2 contraction makes every B chunk use only
// lane-local values: no LDS, no permutes, no selects in the hot loop.

typedef float v2f __attribute__((ext_vector_type(2)));
typedef float v8f __attribute__((ext_vector_type(8)));

#define HID 64
#define NPTS (96 * 96 * 96)  // 884736
#define IMG 128
#define WAVES_PER_BLOCK 8
#define PTS_PER_WAVE 16

__device__ __forceinline__ float fast_tanh(float x) {
  // tanh(x) = 1 - 2/(1 + e^{2x});  2*log2(e) = 2.8853900817779268
  float e = __builtin_amdgcn_exp2f(2.8853900817779268f * x);
  float r = __builtin_amdgcn_rcpf(1.0f + e);
  return 1.0f - 2.0f * r;
}

__device__ __forceinline__ int iclamp(int v, int lo, int hi) {
  return v < lo ? lo : (v > hi ? hi : v);
}

__global__ __launch_bounds__(256) void diffeo_ode_sample_kernel(
    const float* __restrict__ coords,
    const float* __restrict__ W1_0, const float* __restrict__ b1_0,
    const float* __restrict__ W2_0, const float* __restrict__ b2_0,
    const float* __restrict__ W1_1, const float* __restrict__ b1_1,
    const float* __restrict__ W2_1, const float* __restrict__ b2_1,
    const float* __restrict__ vol,
    float* __restrict__ out) {
  const int tid = threadIdx.x;
  const int lane = tid & 31;
  const int wib = tid >> 5;
  const int hmod = lane & 15;
  const bool upper = lane >= 16;
  const bool lower = !upper;
  const int k0 = upper ? 2 : 0;
  const int up8 = upper ? 8 : 0;
  const int jm = hmod & 7;  // duplicated output-row index (rows 0-2 and 8-10)
  const bool jvalid = jm < 3;

  const int wave_id = blockIdx.x * WAVES_PER_BLOCK + wib;
  const int base = wave_id * PTS_PER_WAVE;
  if (base >= NPTS) return;

  // State in WMMA B-layout for y (4x16: K=coord, N=point):
  //   lower lane n: (c0[n], c1[n]);  upper lane n+16: (c2[n], 0-pad)
  const int p = base + hmod;
  float y0 = coords[p * 3 + k0];
  float y1 = lower ? coords[p * 3 + 1] : 0.0f;

  const float h = 0.05f;

#pragma unroll 1
  for (int ode = 0; ode < 2; ++ode) {
    const float* W1 = ode ? W1_1 : W1_0;
    const float* b1 = ode ? b1_1 : b1_0;
    const float* W2 = ode ? W2_1 : W2_0;
    const float* b2 = ode ? b2_1 : b2_0;

    // ---- loop-invariant weights in WMMA A-layouts (registers only) ----
    // Layer 1 A = W1^T tile (M=16 hidden, K=4 coords; K=3 pad -> 0)
    float w1A0[4], w1A1[4];
    v8f bB1v[4];  // bias in C-layout: C[r] = b1[16t + up8 + r] (uniform over N)
#pragma unroll
    for (int t = 0; t < 4; ++t) {
      const int hh = t * 16 + hmod;
      w1A0[t] = W1[k0 * HID + hh];
      w1A1[t] = lower ? W1[1 * HID + hh] : 0.0f;
#pragma unroll
      for (int r = 0; r < 8; ++r) bB1v[t][r] = b1[t * 16 + up8 + r];
    }
    // Layer 2 A = W2^T with custom K-permutation:
    //   chunk kt=(t,q): K0,K1 (lower lanes) -> hidden 16t+2q, 16t+2q+1
    //                   K2,K3 (upper lanes) -> hidden 16t+8+2q, 16t+8+2q+1
    // This matches exactly which hidden values each lane owns after layer 1,
    // so the B operand is lane-local. Output rows duplicated (0-2 and 8-10)
    // so both lane halves read dy from their own acc registers.
    float w2A0[16], w2A1[16];
#pragma unroll
    for (int kt = 0; kt < 16; ++kt) {
      const int t = kt >> 2, q = kt & 3;
      const int kk = 16 * t + up8 + 2 * q;
      w2A0[kt] = jvalid ? W2[kk * 3 + jm] : 0.0f;
      w2A1[kt] = jvalid ? W2[(kk + 1) * 3 + jm] : 0.0f;
    }
    const float b2A0 = b2[k0];  // lower: b2[0], upper: b2[2]
    const float b2A1 = lower ? b2[1] : 0.0f;

    // ---- one MLP evaluation: 20 WMMA + 32 tanh, all lane-local ----
    auto feval = [&](float a0, float a1, float& o0, float& o1) {
      v2f Bin;
      Bin.x = a0;
      Bin.y = a1;
      v8f acc = {0.f, 0.f, 0.f, 0.f, 0.f, 0.f, 0.f, 0.f};
#pragma unroll
      for (int t = 0; t < 4; ++t) {
        v2f Aw;
        Aw.x = w1A0[t];
        Aw.y = w1A1[t];
        // D1 = W1^T x y + b1 : lane n = point, VGPR r = hidden 16t+up8+r
        v8f D = __builtin_amdgcn_wmma_f32_16x16x4_f32(
            false, Aw, false, Bin, (short)0, bB1v[t], false, false);
        float th[8];
#pragma unroll
        for (int r = 0; r < 8; ++r) th[r] = fast_tanh(D[r]);
        // Layer 2: four K=4 chunks; B chunk is (th[2q], th[2q+1]) in every
        // lane thanks to the K-permutation baked into w2A.
#pragma unroll
        for (int q = 0; q < 4; ++q) {
          v2f Bh;
          Bh.x = th[2 * q];
          Bh.y = th[2 * q + 1];
          v2f Aw2;
          Aw2.x = w2A0[4 * t + q];
          Aw2.y = w2A1[4 * t + q];
          acc = __builtin_amdgcn_wmma_f32_16x16x4_f32(
              false, Aw2, false, Bh, (short)0, acc, false, false);
        }
      }
      // Rows 0-2 (lower half) and 8-10 (upper half) both hold dy[point][j].
      o0 = (lower ? acc[0] : acc[2]) + b2A0;
      o1 = lower ? (acc[1] + b2A1) : 0.0f;  // keep K=3 pad exactly zero
    };

    // ---- RK4, 4 fixed steps ----
#pragma unroll 1
    for (int step = 0; step < 4; ++step) {
      float k1a, k1b, k2a, k2b, k3a, k3b, k4a, k4b;
      feval(y0, y1, k1a, k1b);
      feval(y0 + 0.5f * h * k1a, y1 + 0.5f * h * k1b, k2a, k2b);
      feval(y0 + 0.5f * h * k2a, y1 + 0.5f * h * k2b, k3a, k3b);
      feval(y0 + h * k3a, y1 + h * k3b, k4a, k4b);
      y0 += (h / 6.0f) * (k1a + 2.0f * k2a + 2.0f * k3a + k4a);
      y1 += (h / 6.0f) * (k1b + 2.0f * k2b + 2.0f * k3b + k4b);
    }
  }

  // ---- outputs: c1 coords + trilinear sample (lanes 0-15, one point each) ----
  const float c2all = __shfl_xor(y0, 16, 32);  // lower lane gets partner's c2
  if (lane < 16) {
    const int pp = base + lane;
    const float c0 = y0;
    const float c1v = y1;
    const float c2 = c2all;

    float* c1out = out + NPTS;
    c1out[pp * 3 + 0] = c0;
    c1out[pp * 3 + 1] = c1v;
    c1out[pp * 3 + 2] = c2;

    // grid_sample3d: c2->x(W), c1->y(H), c0->z(D); align_corners=False, zeros
    const float x = ((c2 + 1.0f) * (float)IMG - 1.0f) * 0.5f;
    const float yy = ((c1v + 1.0f) * (float)IMG - 1.0f) * 0.5f;
    const float z = ((c0 + 1.0f) * (float)IMG - 1.0f) * 0.5f;
    const float x0f = floorf(x), y0f = floorf(yy), z0f = floorf(z);
    const float tx = x - x0f, ty = yy - y0f, tz = z - z0f;
    const int x0 = (int)x0f, y0i = (int)y0f, z0 = (int)z0f;

    float sacc = 0.0f;
#pragma unroll
    for (int dz = 0; dz < 2; ++dz) {
#pragma unroll
      for (int dy = 0; dy < 2; ++dy) {
#pragma unroll
        for (int dx = 0; dx < 2; ++dx) {
          const int zi = z0 + dz, yi = y0i + dy, xi = x0 + dx;
          const bool valid = (zi >= 0) & (zi < IMG) & (yi >= 0) & (yi < IMG) &
                             (xi >= 0) & (xi < IMG);
          const float w = (dz ? tz : 1.0f - tz) * (dy ? ty : 1.0f - ty) *
                          (dx ? tx : 1.0f - tx);
          const int zc = iclamp(zi, 0, IMG - 1);
          const int yc = iclamp(yi, 0, IMG - 1);
          const int xc = iclamp(xi, 0, IMG - 1);
          const float v = vol[(zc * IMG + yc) * IMG + xc];
          sacc += (valid ? w : 0.0f) * v;
        }
      }
    }
    out[pp] = (sacc - 0.5f) * 2.0f;
  }
}

extern "C" void kernel_launch(void* const* d_in, const int* in_sizes, int n_in,
                              void* d_out, int out_size, void* d_ws,
                              size_t ws_size, hipStream_t stream) {
  (void)in_sizes;
  (void)n_in;
  (void)d_ws;
  (void)ws_size;
  (void)out_size;
  const float* coords = (const float*)d_in[0];
  const float* W1_0 = (const float*)d_in[1];
  const float* b1_0 = (const float*)d_in[2];
  const float* W2_0 = (const float*)d_in[3];
  const float* b2_0 = (const float*)d_in[4];
  const float* W1_1 = (const float*)d_in[5];
  const float* b1_1 = (const float*)d_in[6];
  const float* W2_1 = (const float*)d_in[7];
  const float* b2_1 = (const float*)d_in[8];
  const float* vol = (const float*)d_in[9];

  const int waves = NPTS / PTS_PER_WAVE;       // 55296
  const int blocks = waves / WAVES_PER_BLOCK;  // 6912
  diffeo_ode_sample_kernel<<<blocks, 256, 0, stream>>>(
      coords, W1_0, b1_0, W2_0, b2_0, W1_1, b1_1, W2_1, b2_1, vol,
      (float*)d_out);
}